// VariablesNeighborNorm_15410342658642
// MI455X (gfx1250) — compile-verified
//
#include <hip/hip_runtime.h>
#include <hip/hip_bf16.h>

// Problem constants (match reference)
#define NVARS_K   100000
#define NCLAUSES_K 400000
#define DDIM_K    128
#define ROWPITCH  132          // 129 padded to multiple of 4 for float4 alignment
#define EPS_K     1e-6f

typedef __attribute__((ext_vector_type(2))) float v2f;
typedef __attribute__((ext_vector_type(8))) float v8f;

__device__ __forceinline__ float atomAddF32(float* p, float v) {
    // relaxed agent-scope fp32 atomic -> global_atomic_add_f32
    return __hip_atomic_fetch_add(p, v, __ATOMIC_RELAXED, __HIP_MEMORY_SCOPE_AGENT);
}

// ---------------------------------------------------------------------------
// Phase A: msgs = lit1[cols] * val ; clauses = segment_sum(msgs, rows)
// One wave (32 lanes) per edge; each lane handles 4 of the 128 feature cols.
// ---------------------------------------------------------------------------
__global__ __launch_bounds__(256) void scatter_clauses_kernel(
    const float* __restrict__ vars, const float* __restrict__ vals,
    const int* __restrict__ rows, const int* __restrict__ cols,
    float* __restrict__ clauses, int nnz, int nvars)
{
    int wave = (int)((blockIdx.x * blockDim.x + threadIdx.x) >> 5);
    int lane = (int)(threadIdx.x & 31);
    if (wave >= nnz) return;

    int   row = rows[wave];
    int   col = cols[wave];
    float val = vals[wave];
    int   var = (col >= nvars) ? (col - nvars) : col;   // literals = [vars; vars]

    const float4* src = (const float4*)(vars + (size_t)var * DDIM_K);
    float4 x = src[lane];

    float* dst = clauses + (size_t)row * ROWPITCH + lane * 4;
    atomAddF32(dst + 0, x.x * val);
    atomAddF32(dst + 1, x.y * val);
    atomAddF32(dst + 2, x.z * val);
    atomAddF32(dst + 3, x.w * val);
    if (lane == 0)   // the appended ones-column -> degree accumulation
        atomAddF32(clauses + (size_t)row * ROWPITCH + 128, val);
}

// ---------------------------------------------------------------------------
// Phase B: back = clauses[rows] * val ; varacc = segment_sum(back, col % NVARS)
// (folds lit_a + lit_b directly)
// ---------------------------------------------------------------------------
__global__ __launch_bounds__(256) void gather_scatter_kernel(
    const float* __restrict__ clauses, const float* __restrict__ vals,
    const int* __restrict__ rows, const int* __restrict__ cols,
    float* __restrict__ varacc, int nnz, int nvars)
{
    int wave = (int)((blockIdx.x * blockDim.x + threadIdx.x) >> 5);
    int lane = (int)(threadIdx.x & 31);
    if (wave >= nnz) return;

    int   row = rows[wave];
    int   col = cols[wave];
    float val = vals[wave];
    int   var = (col >= nvars) ? (col - nvars) : col;

    const float* crow = clauses + (size_t)row * ROWPITCH;
    float4 cv = ((const float4*)crow)[lane];

    float* dst = varacc + (size_t)var * ROWPITCH + lane * 4;
    atomAddF32(dst + 0, cv.x * val);
    atomAddF32(dst + 1, cv.y * val);
    atomAddF32(dst + 2, cv.z * val);
    atomAddF32(dst + 3, cv.w * val);
    if (lane == 0)
        atomAddF32(varacc + (size_t)var * ROWPITCH + 128, crow[128] * val);
}

// ---------------------------------------------------------------------------
// Phase C: layer-norm. One wave owns 16 variables (a 16x128 tile of v).
// variance diag comes from G = v * v^T via 32x V_WMMA_F32_16X16X4_F32.
// A operand layout (16x4 f32): lanes 0-15 hold {K=0,K=1}, lanes 16-31 {K=2,K=3},
// M = lane%16.  B (4x16) mirrors it with N = lane%16, so for the symmetric
// Gram product the A and B operands are the SAME register pair.
// ---------------------------------------------------------------------------
__global__ __launch_bounds__(128) void normalize_kernel(
    const float* __restrict__ vars, const float* __restrict__ varacc,
    float* __restrict__ out, int nvars)
{
    __shared__ __align__(16) float tile[4 * 16 * ROWPITCH];  // 33 KB, 4 waves

    int wv   = (int)(threadIdx.x >> 5);
    int lane = (int)(threadIdx.x & 31);
    int base = ((int)blockIdx.x * 4 + wv) * 16;   // first variable of this wave
    float* t = tile + wv * (16 * ROWPITCH);

    // ---- stage v = vars - varacc/max(deg,2) into LDS (coalesced float4) ----
    #pragma unroll
    for (int m = 0; m < 16; ++m) {
        int row = base + m;
        float4 v = make_float4(0.f, 0.f, 0.f, 0.f);
        if (row < nvars) {
            float4 x = ((const float4*)(vars + (size_t)row * DDIM_K))[lane];
            const float* acc = varacc + (size_t)row * ROWPITCH;
            float4 a = ((const float4*)acc)[lane];
            float  md = fmaxf(acc[128], 2.0f);
            float  inv = 1.0f / md;
            v.x = x.x - a.x * inv;
            v.y = x.y - a.y * inv;
            v.z = x.z - a.z * inv;
            v.w = x.w - a.w * inv;
        }
        *(float4*)(t + m * ROWPITCH + lane * 4) = v;
    }
    __syncthreads();

    // ---- Gram diagonal via WMMA: c = sum_k  v[:,k:k+4] x v[:,k:k+4]^T ----
    int off = (lane < 16) ? 0 : 2;
    const float* trow = t + (lane & 15) * ROWPITCH;   // bank-conflict-free (pitch 132)
    v8f c = {};
    #pragma unroll
    for (int k0 = 0; k0 < DDIM_K; k0 += 4) {
        v2f a;
        a.x = trow[k0 + off];
        a.y = trow[k0 + off + 1];
        c = __builtin_amdgcn_wmma_f32_16x16x4_f32(
                /*neg_a=*/false, a, /*neg_b=*/false, a,
                /*c_mod=*/(short)0, c, /*reuse_a=*/false, /*reuse_b=*/false);
    }

    // ---- extract diag: C vgpr r holds M=r (lanes 0-15, N=lane) and M=r+8
    //      (lanes 16-31, N=lane-16).  diag(m,m): m<8 -> c[m]@lane m,
    //      m>=8 -> c[m-8]@lane m+16.
    float scale[16];
    #pragma unroll
    for (int r = 0; r < 8; ++r) {
        float d0 = __shfl(c[r], r, 32);
        float d1 = __shfl(c[r], r + 24, 32);
        scale[r]     = rsqrtf(d0 * (1.0f / 128.0f) + EPS_K);
        scale[r + 8] = rsqrtf(d1 * (1.0f / 128.0f) + EPS_K);
    }

    // ---- write out = v * scale (coalesced float4) ----
    #pragma unroll
    for (int m = 0; m < 16; ++m) {
        int row = base + m;
        if (row < nvars) {
            float4 v = *(const float4*)(t + m * ROWPITCH + lane * 4);
            float  s = scale[m];
            float4 o = make_float4(v.x * s, v.y * s, v.z * s, v.w * s);
            ((float4*)(out + (size_t)row * DDIM_K))[lane] = o;
        }
    }
}

// ---------------------------------------------------------------------------
extern "C" void kernel_launch(void* const* d_in, const int* in_sizes, int n_in,
                              void* d_out, int out_size, void* d_ws, size_t ws_size,
                              hipStream_t stream)
{
    const float* vars = (const float*)d_in[0];
    const float* vals = (const float*)d_in[1];
    const int*   rows = (const int*)d_in[2];
    const int*   cols = (const int*)d_in[3];

    int nvars = in_sizes[0] / DDIM_K;   // 100000
    int nnz   = in_sizes[1];            // 1200000

    float* clauses = (float*)d_ws;                                // 400000*132 f32
    size_t clauseElems = (size_t)NCLAUSES_K * ROWPITCH;
    float* varacc  = clauses + clauseElems;                       // nvars*132 f32
    size_t totalBytes = (clauseElems + (size_t)nvars * ROWPITCH) * sizeof(float);

    hipMemsetAsync(d_ws, 0, totalBytes, stream);                  // graph-capturable

    int edgeBlocks = (nnz + 7) / 8;        // 8 waves of 32 per 256-thread block
    scatter_clauses_kernel<<<edgeBlocks, 256, 0, stream>>>(
        vars, vals, rows, cols, clauses, nnz, nvars);
    gather_scatter_kernel<<<edgeBlocks, 256, 0, stream>>>(
        clauses, vals, rows, cols, varacc, nnz, nvars);

    int normBlocks = (nvars + 63) / 64;    // 4 waves x 16 vars per block
    normalize_kernel<<<normBlocks, 128, 0, stream>>>(
        vars, varacc, (float*)d_out, nvars);
}